// GraphSAGE_32968168964350
// MI455X (gfx1250) — compile-verified
//
#include <hip/hip_runtime.h>
#include <hip/hip_bf16.h>
#include <math.h>

// ---------------------------------------------------------------------------
// GraphSAGE (2-layer, mean aggregation) for MI455X / gfx1250.
// Aggregation: L2-resident atomic scatter (192MB L2 holds all node features).
// GEMMs: exact f32 via V_WMMA_F32_16X16X4_F32; each wave computes TWO 16x16
//        output tiles sharing B fragments (4 WMMAs per B load); A-tiles staged
//        in LDS with the mean-division fused; bias/ReLU fused; branch-free
//        column masking for the 40-wide output layer.
// ---------------------------------------------------------------------------

typedef __attribute__((ext_vector_type(2))) float v2f;
typedef __attribute__((ext_vector_type(8))) float v8f;

#define INC   128   // in/hidden channels
#define OUTC  40

// ---- degree count ---------------------------------------------------------
__global__ void count_deg_kernel(const int* __restrict__ ei, float* __restrict__ cnt, int E) {
    int e = blockIdx.x * blockDim.x + threadIdx.x;
    if (e >= E) return;
    atomicAdd(&cnt[ei[E + e]], 1.0f);
}

// ---- scatter-add of 128-channel messages, 32 threads/edge (float4 each) ---
__global__ void scatter_add_kernel(const int* __restrict__ ei,
                                   const float* __restrict__ feat,
                                   float* __restrict__ agg, int E) {
    int t = blockIdx.x * blockDim.x + threadIdx.x;
    int e = t >> 5;           // edge
    int g = t & 31;           // channel group of 4
    if (e >= E) return;
    int s = ei[e];
    int d = ei[E + e];
    const float4 v = *(const float4*)(feat + (size_t)s * 128 + g * 4);
    float* p = agg + (size_t)d * 128 + g * 4;
    atomicAdd(p + 0, v.x);
    atomicAdd(p + 1, v.y);
    atomicAdd(p + 2, v.z);
    atomicAdd(p + 3, v.w);
}

// ---- fused SAGE GEMM: out = act( (agg/deg)@Wl + Ax@Wr + b ) ---------------
// 32-row M-block per workgroup; wave w owns column tile w and computes two
// 16x16 tiles (rows 0-15 and 16-31 of the block) reusing one B fragment.
template <int COLT, bool RELU, int VALID_N, int W_STRIDE, int OUT_STRIDE>
__global__ __launch_bounds__(COLT * 32) void sage_gemm_kernel(
    const float* __restrict__ Aagg, const float* __restrict__ cnt,
    const float* __restrict__ Ax,
    const float* __restrict__ Wl, const float* __restrict__ Wr,
    const float* __restrict__ bias, float* __restrict__ out) {
    constexpr int K = 128;
    constexpr bool FULL = (VALID_N == COLT * 16);   // no column masking needed
    __shared__ float lA0[32 * K];   // mean-aggregated neighbor features (scaled)
    __shared__ float lA1[32 * K];   // root features

    const int tid = threadIdx.x;
    const size_t rowBase = (size_t)blockIdx.x * 32;

    // Stage both A tiles to LDS; fuse the 1/max(deg,1) division here.
    for (int i = tid; i < 32 * K / 4; i += COLT * 32) {
        const int r  = i >> 5;                   // 32 float4 per 128-wide row
        const int c4 = i & 31;
        const float invc = 1.0f / fmaxf(cnt[rowBase + r], 1.0f);
        float4 va = ((const float4*)(Aagg + (rowBase + r) * K))[c4];
        va.x *= invc; va.y *= invc; va.z *= invc; va.w *= invc;
        ((float4*)lA0)[i] = va;
        ((float4*)lA1)[i] = ((const float4*)(Ax + (rowBase + r) * K))[c4];
    }
    __syncthreads();

    const int wave = tid >> 5;
    const int lane = tid & 31;
    const int half = lane >> 4;         // 0: lanes 0-15, 1: lanes 16-31
    const int l16  = lane & 15;
    const int ncol = wave * 16 + l16;   // output column
    // Branch-free masking for partial column tiles (layer 2, 40 cols):
    const int   ncolL = FULL ? ncol : (ncol < VALID_N ? ncol : VALID_N - 1);
    const float mask  = (FULL || ncol < VALID_N) ? 1.0f : 0.0f;

    const float bval = FULL ? bias[ncol] : bias[ncolL] * mask;
    v8f acc0 = {bval, bval, bval, bval, bval, bval, bval, bval};
    v8f acc1 = acc0;

    const int kh = half * 2;            // A/B frag: lanes 16-31 carry K+2,K+3
    const float* __restrict__ a0lo = lA0 + l16 * K;
    const float* __restrict__ a0hi = lA0 + (16 + l16) * K;
    const float* __restrict__ a1lo = lA1 + l16 * K;
    const float* __restrict__ a1hi = lA1 + (16 + l16) * K;

#pragma unroll
    for (int kb = 0; kb < K; kb += 4) {
        v2f b0, b1, a;
        if constexpr (FULL) {
            b0.x = Wl[(kb + kh) * W_STRIDE + ncol];
            b0.y = Wl[(kb + kh + 1) * W_STRIDE + ncol];
            b1.x = Wr[(kb + kh) * W_STRIDE + ncol];
            b1.y = Wr[(kb + kh + 1) * W_STRIDE + ncol];
        } else {
            b0.x = Wl[(kb + kh) * W_STRIDE + ncolL] * mask;
            b0.y = Wl[(kb + kh + 1) * W_STRIDE + ncolL] * mask;
            b1.x = Wr[(kb + kh) * W_STRIDE + ncolL] * mask;
            b1.y = Wr[(kb + kh + 1) * W_STRIDE + ncolL] * mask;
        }
        a.x = a0lo[kb + kh]; a.y = a0lo[kb + kh + 1];
        acc0 = __builtin_amdgcn_wmma_f32_16x16x4_f32(false, a, false, b0,
                                                     (short)0, acc0, false, false);
        a.x = a1lo[kb + kh]; a.y = a1lo[kb + kh + 1];
        acc0 = __builtin_amdgcn_wmma_f32_16x16x4_f32(false, a, false, b1,
                                                     (short)0, acc0, false, false);
        a.x = a0hi[kb + kh]; a.y = a0hi[kb + kh + 1];
        acc1 = __builtin_amdgcn_wmma_f32_16x16x4_f32(false, a, false, b0,
                                                     (short)0, acc1, false, false);
        a.x = a1hi[kb + kh]; a.y = a1hi[kb + kh + 1];
        acc1 = __builtin_amdgcn_wmma_f32_16x16x4_f32(false, a, false, b1,
                                                     (short)0, acc1, false, false);
    }

#pragma unroll
    for (int i = 0; i < 8; ++i) {
        const int m = i + half * 8;     // C layout: VGPR i -> M=i (lo) / M=8+i (hi)
        float v0 = acc0[i];
        float v1 = acc1[i];
        if (RELU) { v0 = fmaxf(v0, 0.0f); v1 = fmaxf(v1, 0.0f); }
        if (FULL || ncol < VALID_N) {
            out[(rowBase + m) * OUT_STRIDE + ncol]      = v0;
            out[(rowBase + 16 + m) * OUT_STRIDE + ncol] = v1;
        }
    }
}

// ---- in-place log-softmax over 40 classes ---------------------------------
__global__ void logsoftmax_kernel(float* __restrict__ out, int N) {
    int n = blockIdx.x * blockDim.x + threadIdx.x;
    if (n >= N) return;
    float* p = out + (size_t)n * OUTC;
    float m = -INFINITY;
#pragma unroll
    for (int i = 0; i < OUTC; ++i) m = fmaxf(m, p[i]);
    float s = 0.0f;
#pragma unroll
    for (int i = 0; i < OUTC; ++i) s += expf(p[i] - m);
    const float lse = m + logf(s);
#pragma unroll
    for (int i = 0; i < OUTC; ++i) p[i] -= lse;
}

extern "C" void kernel_launch(void* const* d_in, const int* in_sizes, int n_in,
                              void* d_out, int out_size, void* d_ws, size_t ws_size,
                              hipStream_t stream) {
    const float* x   = (const float*)d_in[0];
    const int*   ei  = (const int*)d_in[1];   // [2, E]: row 0 = src, row 1 = dst
    const float* W1l = (const float*)d_in[2];
    const float* W1r = (const float*)d_in[3];
    const float* b1  = (const float*)d_in[4];
    const float* W2l = (const float*)d_in[5];
    const float* W2r = (const float*)d_in[6];
    const float* b2  = (const float*)d_in[7];
    float* out = (float*)d_out;

    const int N = in_sizes[0] / INC;      // 100000
    const int E = in_sizes[1] / 2;        // 1600000

    // Workspace: agg [N*128] | cnt [N] | h [N*128]
    float* agg = (float*)d_ws;
    float* cnt = agg + (size_t)N * INC;
    float* h   = cnt + N;

    const int mb = N / 32;                // 3125 exact 32-row M blocks

    // ---- layer 1 ----
    hipMemsetAsync(agg, 0, ((size_t)N * INC + N) * sizeof(float), stream);
    count_deg_kernel<<<(E + 255) / 256, 256, 0, stream>>>(ei, cnt, E);
    scatter_add_kernel<<<((size_t)E * 32 + 255) / 256, 256, 0, stream>>>(ei, x, agg, E);
    sage_gemm_kernel<8, true, 128, 128, 128><<<mb, 256, 0, stream>>>(
        agg, cnt, x, W1l, W1r, b1, h);

    // ---- layer 2 (degrees unchanged; re-zero accumulator only) ----
    hipMemsetAsync(agg, 0, (size_t)N * INC * sizeof(float), stream);
    scatter_add_kernel<<<((size_t)E * 32 + 255) / 256, 256, 0, stream>>>(ei, h, agg, E);
    sage_gemm_kernel<3, false, OUTC, OUTC, OUTC><<<mb, 96, 0, stream>>>(
        agg, cnt, h, W2l, W2r, b2, out);

    // ---- log-softmax ----
    logsoftmax_kernel<<<(N + 255) / 256, 256, 0, stream>>>(out, N);
}